// GraphConvNet_71159018160289
// MI455X (gfx1250) — compile-verified
//
#include <hip/hip_runtime.h>
#include <hip/hip_bf16.h>

typedef __attribute__((ext_vector_type(2))) float v2f;
typedef __attribute__((ext_vector_type(8))) float v8f;

#define H128 128
#define LDSW 132   // padded LDS row stride (floats): conflict-free b64 fragment reads

template<int ACT> __device__ __forceinline__ float act_apply(float x) {
    if (ACT == 1) return x > 0.f ? x : 0.01f * x;          // LeakyReLU(0.01)
    if (ACT == 2) return 1.f / (1.f + expf(-x));           // sigmoid
    return x;
}

// Y[nrows x 128] = act(X[nrows x 128] @ W[128 x 128] + b) (optionally * mulsrc, dual-store)
// Block = 256 threads = 8 waves; each wave owns a 16-row stripe x all 128 cols (8 WMMA tiles).
template<int ACT, bool MUL, bool DUAL>
__global__ __launch_bounds__(256)
void gemm128_wmma(const float* __restrict__ X, const float* __restrict__ W,
                  const float* __restrict__ bias, const float* __restrict__ mulsrc,
                  float* __restrict__ Y, float* __restrict__ Y2, int nrows)
{
    extern __shared__ float lds[];
    float* xs = lds;               // [128][LDSW]  X tile, row-major (m, k)
    float* wt = lds + H128 * LDSW; // [128][LDSW]  W transposed (n, k)

    const int tid    = threadIdx.x;
    const int rowblk = blockIdx.x * 128;

    // Stage W transposed: thread reads W[k][n4..n4+3] coalesced, scatters to wt[n][k]
    for (int i = tid; i < 128 * 32; i += 256) {
        const int k  = i >> 5;
        const int n4 = (i & 31) << 2;
        const float4 v = *(const float4*)(W + (k << 7) + n4);
        wt[(n4 + 0) * LDSW + k] = v.x;
        wt[(n4 + 1) * LDSW + k] = v.y;
        wt[(n4 + 2) * LDSW + k] = v.z;
        wt[(n4 + 3) * LDSW + k] = v.w;
    }
    // Stage X tile (coalesced b128 loads), zero-pad rows past nrows
    for (int i = tid; i < 128 * 32; i += 256) {
        const int m   = i >> 5;
        const int k4  = (i & 31) << 2;
        const int row = rowblk + m;
        float4 v = make_float4(0.f, 0.f, 0.f, 0.f);
        if (row < nrows) v = *(const float4*)(X + (size_t)row * H128 + k4);
        *(float4*)(xs + m * LDSW + k4) = v;
    }
    __syncthreads();

    const int wave = tid >> 5;
    const int lane = tid & 31;
    const int n    = lane & 15;       // col within 16-wide tile (B/C layout)
    const int half = lane >> 4;       // lane group selects K pair (A/B) / M+8 (C)
    const int kb   = half << 1;       // A frag: v0=K(kb), v1=K(kb+1)
    const int mrow = wave * 16 + n;   // A-matrix row served by this lane

    v8f acc[8] = {};

    const float* ax = xs + mrow * LDSW + kb;
    #pragma unroll 4
    for (int k0 = 0; k0 < 128; k0 += 4) {
        const v2f a = *(const v2f*)(ax + k0);                     // A 16x4 fragment
        #pragma unroll
        for (int t = 0; t < 8; ++t) {
            const v2f b = *(const v2f*)(wt + (t * 16 + n) * LDSW + kb + k0); // B 4x16
            acc[t] = __builtin_amdgcn_wmma_f32_16x16x4_f32(
                false, a, false, b, (short)0, acc[t], false, false);
        }
    }

    // C/D layout: VGPR r -> M = r + 8*half, N = lane&15
    #pragma unroll
    for (int r = 0; r < 8; ++r) {
        const int row = rowblk + wave * 16 + half * 8 + r;
        if (row < nrows) {
            #pragma unroll
            for (int t = 0; t < 8; ++t) {
                const int col = t * 16 + n;
                float v = acc[t][r] + bias[col];
                v = act_apply<ACT>(v);
                if (MUL) v *= mulsrc[(size_t)row * H128 + col];
                Y[(size_t)row * H128 + col] = v;
                if (DUAL) Y2[(size_t)row * H128 + col] = v;
            }
        }
    }
}

__global__ void zero_f32(float* __restrict__ p, int n) {
    int i = blockIdx.x * blockDim.x + threadIdx.x;
    if (i < n) p[i] = 0.f;
}

__global__ void deg_count(const int* __restrict__ dst, float* __restrict__ deg, int E) {
    int e = blockIdx.x * blockDim.x + threadIdx.x;
    if (e < E) atomicAdd(deg + dst[e], 1.0f);
}

__global__ void inv_finalize(float* __restrict__ deg, int n) {
    int i = blockIdx.x * blockDim.x + threadIdx.x;
    if (i < n) deg[i] = 1.0f / (deg[i] + 1.0f);
}

// out[dst] += h[src]; one thread per (edge, 4-col chunk): 32 threads cover one 512B row
__global__ void scatter_add(const float* __restrict__ h, const int* __restrict__ dst,
                            const int* __restrict__ src, float* __restrict__ out, int E) {
    int t = blockIdx.x * blockDim.x + threadIdx.x;
    if (t >= E * 32) return;
    const int e  = t >> 5;
    const int c4 = (t & 31) << 2;
    const int s = src[e];
    const int d = dst[e];
    const float4 v = *(const float4*)(h + (size_t)s * H128 + c4);
    float* o = out + (size_t)d * H128 + c4;
    atomicAdd(o + 0, v.x);
    atomicAdd(o + 1, v.y);
    atomicAdd(o + 2, v.z);
    atomicAdd(o + 3, v.w);
}

__global__ void scale_rows(float* __restrict__ x, const float* __restrict__ inv, int N) {
    int t = blockIdx.x * blockDim.x + threadIdx.x;
    if (t >= N * 32) return;
    const int i  = t >> 5;
    const int c4 = (t & 31) << 2;
    float4 v = *(float4*)(x + (size_t)i * H128 + c4);
    const float s = inv[i];
    v.x *= s; v.y *= s; v.z *= s; v.w *= s;
    *(float4*)(x + (size_t)i * H128 + c4) = v;
}

// Per-graph mean over `apg` contiguous atoms, then [1x128]@[128xC] + b
__global__ void pool_final(const float* __restrict__ x, const float* __restrict__ finW,
                           const float* __restrict__ finb, float* __restrict__ out,
                           int apg, float inv_apg, int C) {
    __shared__ float m[H128];
    const int g = blockIdx.x;
    const int c = threadIdx.x; // 128 threads
    float s = 0.f;
    const float* base = x + (size_t)g * apg * H128 + c;
    for (int j = 0; j < apg; ++j) s += base[(size_t)j * H128];
    m[c] = s * inv_apg;
    __syncthreads();
    if (c < C) {
        float acc = finb[c];
        #pragma unroll 8
        for (int h = 0; h < H128; ++h) acc += m[h] * finW[h * C + c];
        out[(size_t)g * C + c] = acc;
    }
}

extern "C" void kernel_launch(void* const* d_in, const int* in_sizes, int n_in,
                              void* d_out, int out_size, void* d_ws, size_t ws_size,
                              hipStream_t stream) {
    const float* node_attr = (const float*)d_in[0];
    const int*   ei        = (const int*)d_in[1];   // JAX default x64 off -> int32
    const float* enc_W = (const float*)d_in[3];
    const float* enc_b = (const float*)d_in[4];
    const float* gcn_W = (const float*)d_in[5];
    const float* gcn_b = (const float*)d_in[6];
    const float* r_W1  = (const float*)d_in[7];
    const float* r_b1  = (const float*)d_in[8];
    const float* r_W2  = (const float*)d_in[9];
    const float* r_b2  = (const float*)d_in[10];
    const float* s_W1  = (const float*)d_in[11];
    const float* s_b1  = (const float*)d_in[12];
    const float* s_W2  = (const float*)d_in[13];
    const float* s_b2  = (const float*)d_in[14];
    const float* fin_W = (const float*)d_in[15];
    const float* fin_b = (const float*)d_in[16];

    const int N = in_sizes[0] / H128;
    const int E = in_sizes[1] / 2;
    const int G = in_sizes[2];
    const int L = in_sizes[5] / (H128 * H128);
    const int C = in_sizes[16];
    const int apg = N / G;
    const int* e_dst = ei;
    const int* e_src = ei + E;

    float* bufX = (float*)d_ws;
    float* bufH = bufX + (size_t)N * H128;
    float* bufR = bufH + (size_t)N * H128;
    float* dinv = bufR + (size_t)N * H128;

    const int SMEM = 2 * H128 * LDSW * (int)sizeof(float);
    const int gemm_blocks = (N + 127) / 128;

    // degree -> inv_deg
    zero_f32<<<(N + 255) / 256, 256, 0, stream>>>(dinv, N);
    deg_count<<<(E + 255) / 256, 256, 0, stream>>>(e_dst, dinv, E);
    inv_finalize<<<(N + 255) / 256, 256, 0, stream>>>(dinv, N);

    // encoder: x = node_attr @ enc_W + enc_b
    gemm128_wmma<0, false, false><<<gemm_blocks, 256, SMEM, stream>>>(
        node_attr, enc_W, enc_b, nullptr, bufX, nullptr, N);

    // GCN layers: h = x@W+b (dual-store into bufH and bufX); bufX[dst] += bufH[src]; x *= inv_deg
    for (int l = 0; l < L; ++l) {
        gemm128_wmma<0, false, true><<<gemm_blocks, 256, SMEM, stream>>>(
            bufX, gcn_W + (size_t)l * H128 * H128, gcn_b + (size_t)l * H128,
            nullptr, bufH, bufX, N);
        scatter_add<<<(E * 32 + 255) / 256, 256, 0, stream>>>(bufH, e_dst, e_src, bufX, E);
        scale_rows<<<(N * 32 + 255) / 256, 256, 0, stream>>>(bufX, dinv, N);
    }

    // Gate MLP: r = leaky(leaky(x@rW1+b)@rW2+b); x = r * sigmoid(leaky(x@sW1+b)@sW2+b)
    gemm128_wmma<1, false, false><<<gemm_blocks, 256, SMEM, stream>>>(
        bufX, r_W1, r_b1, nullptr, bufH, nullptr, N);
    gemm128_wmma<1, false, false><<<gemm_blocks, 256, SMEM, stream>>>(
        bufH, r_W2, r_b2, nullptr, bufR, nullptr, N);
    gemm128_wmma<1, false, false><<<gemm_blocks, 256, SMEM, stream>>>(
        bufX, s_W1, s_b1, nullptr, bufH, nullptr, N);
    gemm128_wmma<2, true, false><<<gemm_blocks, 256, SMEM, stream>>>(
        bufH, s_W2, s_b2, bufR, bufX, nullptr, N);

    // mean-pool per graph + final [128 x C] GEMM
    pool_final<<<G, H128, 0, stream>>>(bufX, fin_W, fin_b, (float*)d_out,
                                       apg, 1.0f / (float)apg, C);
}